// GCNIIConv_6150393168666
// MI455X (gfx1250) — compile-verified
//
#include <hip/hip_runtime.h>

typedef float v2f __attribute__((ext_vector_type(2)));
typedef float v4f __attribute__((ext_vector_type(4)));
typedef float v8f __attribute__((ext_vector_type(8)));

#define CCH 128
#define BETA_F 0.6931471805599453f          // log(2)
#define C1_F   0.15342640972002733f         // (1-ALPHA)*(1-BETA) == ALPHA*(1-BETA), ALPHA=0.5

// ---------------- degree / norm ----------------
__global__ void k_deg_init(float* __restrict__ deg, int n) {
    int i = blockIdx.x * blockDim.x + threadIdx.x;
    if (i < n) deg[i] = 1.0f;                       // self-loop contribution
}

__global__ void k_deg_scatter(const int* __restrict__ dst, float* __restrict__ deg, int E) {
    int i = blockIdx.x * blockDim.x + threadIdx.x;
    if (i < E) atomicAdd(&deg[dst[i]], 1.0f);
}

__global__ void k_deg_rsqrt(float* __restrict__ deg, int n) {
    int i = blockIdx.x * blockDim.x + threadIdx.x;
    if (i < n) deg[i] = rsqrtf(deg[i]);             // deg >= 1 always (self-loops)
}

// ---------------- SpMM: h = D^-1/2 (A+I) D^-1/2 x ----------------
// self-loop term: h[i] = dis[i]^2 * x[i]   (also fully initializes h scratch)
__global__ void k_h_init(const float* __restrict__ x, const float* __restrict__ dis,
                         float* __restrict__ h, int n4) {
    int i = blockIdx.x * blockDim.x + threadIdx.x;  // over N*C/4 float4's
    if (i < n4) {
        int node = i >> 5;                          // 32 float4's per node (C=128)
        float d = dis[node];
        float s = d * d;
        ((v4f*)h)[i] = ((const v4f*)x)[i] * s;
    }
}

// one wave per edge, 4 channels per lane: float4 gather + 4 fp32 global atomics
__global__ void k_edge_scatter(const float* __restrict__ x, const float* __restrict__ dis,
                               const int* __restrict__ src, const int* __restrict__ dst,
                               float* __restrict__ h, int E) {
    int e    = blockIdx.x * 8 + (threadIdx.x >> 5);
    int lane = threadIdx.x & 31;
    if (e >= E) return;
    int s = src[e];
    int d = dst[e];
    float nrm = dis[s] * dis[d];
    v4f xv = *(const v4f*)(x + (size_t)s * CCH + lane * 4);
    float* hp = h + (size_t)d * CCH + lane * 4;
    atomicAdd(hp + 0, nrm * xv.x);
    atomicAdd(hp + 1, nrm * xv.y);
    atomicAdd(hp + 2, nrm * xv.z);
    atomicAdd(hp + 3, nrm * xv.w);
}

// ---------------- fused  out = relu(c1*h + h@(B*W1) + c1*x0 + x0@(B*W2)) ----------------
// 256 threads = 8 waves; block owns 16 rows, wave w owns cols [16w, 16w+16).
// fp32 WMMA 16x16x4, K accumulated in steps of 4 (32 wmma per GEMM term).
__global__ void __launch_bounds__(256) k_gcn2_wmma(
        const float* __restrict__ h,  const float* __restrict__ x0,
        const float* __restrict__ w1, const float* __restrict__ w2,
        float* __restrict__ out) {
    extern __shared__ float lw[];                   // 2 * 128*128 f32 = 128 KB (BETA-prescaled)
    const int tid = threadIdx.x;

    // stage BETA*W1, BETA*W2 into LDS (folds the beta scale into the B operand)
    for (int i = tid * 4; i < CCH * CCH; i += 256 * 4) {
        *(v4f*)(lw + i)             = *(const v4f*)(w1 + i) * BETA_F;
        *(v4f*)(lw + CCH * CCH + i) = *(const v4f*)(w2 + i) * BETA_F;
    }

    const int lane   = tid & 31;
    const int wave   = tid >> 5;
    const int lanelo = lane & 15;
    const int hi     = lane >> 4;                   // 0: K=k0..k0+1 | 1: K=k0+2..k0+3
    const int row0   = blockIdx.x * 16;
    const int n0     = wave * 16;

    // C/D layout: VGPR v -> row (row0 + hi*8 + v), col (n0 + lanelo)
    const int rbase = row0 + hi * 8;
    const int ncol  = n0 + lanelo;

    // Seed accumulator with elementwise terms BEFORE the barrier: these global
    // loads overlap the LDS weight staging and also warm the WGP$ with the
    // A-strip rows the k-loop re-reads. (Both coefficients equal for ALPHA=0.5.)
    v8f acc = {};
#pragma unroll
    for (int v = 0; v < 8; ++v) {
        size_t off = (size_t)(rbase + v) * CCH + ncol;
        acc[v] = C1_F * (h[off] + x0[off]);
    }

    __syncthreads();                                // weights visible in LDS

    // A operand pointers: lane holds row (row0+lanelo), K pair starting at k0 + 2*hi
    const float* ap1 = h  + (size_t)(row0 + lanelo) * CCH + hi * 2;
    const float* ap2 = x0 + (size_t)(row0 + lanelo) * CCH + hi * 2;
    // B operand pointers in LDS: row K = k0 + 2*hi, col ncol
    const float* bp1 = lw + hi * 2 * CCH + ncol;
    const float* bp2 = lw + CCH * CCH + hi * 2 * CCH + ncol;

#pragma unroll
    for (int k0 = 0; k0 < CCH; k0 += 4) {           // h @ (BETA*W1)
        v2f a = *(const v2f*)(ap1 + k0);
        v2f b = { bp1[k0 * CCH], bp1[k0 * CCH + CCH] };
        acc = __builtin_amdgcn_wmma_f32_16x16x4_f32(false, a, false, b,
                                                    (short)0, acc, false, false);
    }
#pragma unroll
    for (int k0 = 0; k0 < CCH; k0 += 4) {           // x0 @ (BETA*W2)
        v2f a = *(const v2f*)(ap2 + k0);
        v2f b = { bp2[k0 * CCH], bp2[k0 * CCH + CCH] };
        acc = __builtin_amdgcn_wmma_f32_16x16x4_f32(false, a, false, b,
                                                    (short)0, acc, false, false);
    }

    // ReLU + non-temporal store: out is write-only, keep it out of L2 so the
    // x / h working set stays resident for the gather/atomic traffic.
#pragma unroll
    for (int v = 0; v < 8; ++v) {
        size_t off = (size_t)(rbase + v) * CCH + ncol;
        __builtin_nontemporal_store(fmaxf(acc[v], 0.0f), out + off);
    }
}

extern "C" void kernel_launch(void* const* d_in, const int* in_sizes, int n_in,
                              void* d_out, int out_size, void* d_ws, size_t ws_size,
                              hipStream_t stream) {
    (void)n_in; (void)out_size; (void)ws_size;
    const float* x   = (const float*)d_in[0];
    const float* x0  = (const float*)d_in[1];
    const float* w1  = (const float*)d_in[2];
    const float* w2  = (const float*)d_in[3];
    const int*   ei  = (const int*)d_in[4];
    float* out = (float*)d_out;

    const int N = in_sizes[0] / CCH;                // 100000 (multiple of 16)
    const int E = in_sizes[4] / 2;                  // 800000
    const int* srcv = ei;                           // edge_index[0]
    const int* dstv = ei + E;                       // edge_index[1]

    float* deg  = (float*)d_ws;                     // N floats (becomes dis in place)
    float* hbuf = deg + N;                          // N*CCH floats, 16B-aligned

    k_deg_init   <<<(N + 255) / 256, 256, 0, stream>>>(deg, N);
    k_deg_scatter<<<(E + 255) / 256, 256, 0, stream>>>(dstv, deg, E);
    k_deg_rsqrt  <<<(N + 255) / 256, 256, 0, stream>>>(deg, N);

    const int n4 = N * (CCH / 4);
    k_h_init      <<<(n4 + 255) / 256, 256, 0, stream>>>(x, deg, hbuf, n4);
    k_edge_scatter<<<(E + 7) / 8, 256, 0, stream>>>(x, deg, srcv, dstv, hbuf, E);

    k_gcn2_wmma<<<N / 16, 256, 2 * CCH * CCH * sizeof(float), stream>>>(
        hbuf, x0, w1, w2, out);
}